// GradTTS_23003844838043
// MI455X (gfx1250) — compile-verified
//
#include <hip/hip_runtime.h>
#include <hip/hip_bf16.h>
#include <math.h>

// Problem sizes (fixed by the reference)
#define BB    32
#define TXDIM 256
#define TYDIM 1728
#define NMEL  80
#define NEGF  (-1.0e9f)
#define LOG2PI_F 1.8378770664093453f
#define PRIOR_CONST (-0.5f * LOG2PI_F * (float)NMEL)   // -0.5*log(2pi)*NMEL

typedef float v2f __attribute__((ext_vector_type(2)));
typedef float v8f __attribute__((ext_vector_type(8)));

// ---- workspace layout (bytes) ----
// lp   : [B][Ty][Tx] f32  (x contiguous -> DP reads a contiguous 1KB column)
// dirs : [B][Ty][32] u8   (bit k of byte `lane` = came_from_below at x=lane*8+k)
// xi   : [B][Ty] i32      (path index per mel frame; -1 when t >= y_len)
// ysq  : [B][Ty] f32, msq : [B][Tx] f32, acc : 2 f32 (dur_acc, prior_acc)
#define LP_OFF   ((size_t)0)
#define DIRS_OFF (LP_OFF  + (size_t)BB*TYDIM*TXDIM*4)
#define XI_OFF   (DIRS_OFF + (size_t)BB*TYDIM*32)
#define YSQ_OFF  (XI_OFF  + (size_t)BB*TYDIM*4)
#define MSQ_OFF  (YSQ_OFF + (size_t)BB*TYDIM*4)
#define ACC_OFF  (MSQ_OFF + (size_t)BB*TXDIM*4)

// ---------------------------------------------------------------------------
// K0: -0.5*||y_t||^2 and -0.5*||mu_x||^2 row norms; zero the loss accumulators
// ---------------------------------------------------------------------------
__global__ void __launch_bounds__(256) gtts_sq_kernel(
    const float* __restrict__ mu_x, const float* __restrict__ y,
    float* __restrict__ ysq, float* __restrict__ msq, float* __restrict__ acc)
{
  int i = blockIdx.x * 256 + threadIdx.x;
  if (i == 0) { acc[0] = 0.0f; acc[1] = 0.0f; }
  const int NY = BB * TYDIM;
  if (i < NY) {
    const float* p = y + (size_t)i * NMEL;
    float s = 0.0f;
#pragma unroll
    for (int m = 0; m < NMEL; ++m) s += p[m] * p[m];
    ysq[i] = -0.5f * s;
  } else if (i < NY + BB * TXDIM) {
    int j = i - NY;
    const float* p = mu_x + (size_t)j * NMEL;
    float s = 0.0f;
#pragma unroll
    for (int m = 0; m < NMEL; ++m) s += p[m] * p[m];
    msq[j] = -0.5f * s;
  }
}

// ---------------------------------------------------------------------------
// K1: log-prior via fp32 WMMA, register-blocked 2x2 (one wave -> 32t x 32x).
//   Each A/B fragment feeds two WMMAs -> halves L2 fragment traffic vs 1x1.
//   lp[b][t][x] = ysq[b,t] + sum_m y[b,t,m]*mu_x[b,x,m] + msq[b,x] + const,
//   masked to NEG outside (x < x_len, t < y_len).
// V_WMMA_F32_16X16X4_F32 layouts (ISA 7.12.2):
//   A 16x4 : lane L holds row M=L&15; VGPR r holds K = r + 2*(L>>4)
//   B  4x16: assumed symmetric: lane L holds col N=L&15; VGPR r = K r+2*(L>>4)
//   C/D    : lane L col N=L&15; VGPR r holds row M = r + 8*(L>>4)
// ---------------------------------------------------------------------------
__global__ void __launch_bounds__(32) gtts_logprior_wmma_kernel(
    const float* __restrict__ mu_x, const float* __restrict__ y,
    const float* __restrict__ ysq, const float* __restrict__ msq,
    const int* __restrict__ xlen, const int* __restrict__ ylen,
    float* __restrict__ lp)
{
  const int x0 = blockIdx.x * 32;
  const int t0 = blockIdx.y * 32;
  const int b  = blockIdx.z;
  const int lane = threadIdx.x & 31;
  const int half = lane >> 4;
  const int l15  = lane & 15;

  // A rows come from y (M = mel frame), B cols from mu_x (N = text position)
  const float* arow0 = y    + ((size_t)b * TYDIM + (t0 + l15)) * NMEL + 2 * half;
  const float* arow1 = arow0 + (size_t)16 * NMEL;
  const float* bcol0 = mu_x + ((size_t)b * TXDIM + (x0 + l15)) * NMEL + 2 * half;
  const float* bcol1 = bcol0 + (size_t)16 * NMEL;

  v8f c00 = {}, c01 = {}, c10 = {}, c11 = {};
#pragma unroll
  for (int k = 0; k < NMEL; k += 4) {
    v2f a0 = *(const v2f*)(arow0 + k);   // K = k+2*half .. +1   (8B aligned)
    v2f a1 = *(const v2f*)(arow1 + k);
    v2f b0 = *(const v2f*)(bcol0 + k);
    v2f b1 = *(const v2f*)(bcol1 + k);
    c00 = __builtin_amdgcn_wmma_f32_16x16x4_f32(false, a0, false, b0, (short)0, c00, false, false);
    c01 = __builtin_amdgcn_wmma_f32_16x16x4_f32(false, a0, false, b1, (short)0, c01, false, false);
    c10 = __builtin_amdgcn_wmma_f32_16x16x4_f32(false, a1, false, b0, (short)0, c10, false, false);
    c11 = __builtin_amdgcn_wmma_f32_16x16x4_f32(false, a1, false, b1, (short)0, c11, false, false);
  }

  const int xl = xlen[b], yl = ylen[b];
  const float ms0 = msq[b * TXDIM + x0 + l15];
  const float ms1 = msq[b * TXDIM + x0 + 16 + l15];
  const bool xin0 = (x0 + l15) < xl;
  const bool xin1 = (x0 + 16 + l15) < xl;
  float* lpb = lp + (size_t)b * TYDIM * TXDIM;
#pragma unroll
  for (int r = 0; r < 8; ++r) {
    const int ta = t0 + r + 8 * half;          // rows of c00/c01
    const int tb = ta + 16;                    // rows of c10/c11
    const float ysa = ysq[b * TYDIM + ta];
    const float ysb = ysq[b * TYDIM + tb];
    const bool ina = (ta < yl);
    const bool inb = (tb < yl);
    lpb[(size_t)ta * TXDIM + (x0 + l15)]      = (xin0 && ina) ? (c00[r] + ysa + ms0 + PRIOR_CONST) : NEGF;
    lpb[(size_t)ta * TXDIM + (x0 + 16 + l15)] = (xin1 && ina) ? (c01[r] + ysa + ms1 + PRIOR_CONST) : NEGF;
    lpb[(size_t)tb * TXDIM + (x0 + l15)]      = (xin0 && inb) ? (c10[r] + ysb + ms0 + PRIOR_CONST) : NEGF;
    lpb[(size_t)tb * TXDIM + (x0 + 16 + l15)] = (xin1 && inb) ? (c11[r] + ysb + ms1 + PRIOR_CONST) : NEGF;
  }
}

// ---------------------------------------------------------------------------
// K2: Viterbi forward. One wave32 per batch; lane owns x = lane*8..lane*8+7.
// Sequential over t; 4-column register pipeline + 16-column prefetch
// (global_prefetch_b8) hide L2 latency on the serial dependence chain.
// ---------------------------------------------------------------------------
__global__ void __launch_bounds__(32) gtts_viterbi_fwd_kernel(
    const float* __restrict__ lp, unsigned char* __restrict__ dirs)
{
  const int b = blockIdx.x;
  const int lane = threadIdx.x & 31;
  const float* lpb = lp + (size_t)b * TYDIM * TXDIM;

  float v[8];
#pragma unroll
  for (int k = 0; k < 8; ++k) v[k] = NEGF;
  if (lane == 0) v[0] = lpb[0];          // t=0: only x=0 reachable

  float4 blo[4], bhi[4];
#pragma unroll
  for (int d = 0; d < 4; ++d) {          // prime pipeline with t=1..4
    const float4* p = (const float4*)(lpb + (size_t)(1 + d) * TXDIM + lane * 8);
    blo[d] = p[0]; bhi[d] = p[1];
  }

  unsigned char* db = dirs + (size_t)b * TYDIM * 32;
  for (int t0 = 1; t0 < TYDIM; t0 += 4) {
    // warm the near cache well ahead of the register pipeline
    if (t0 + 16 < TYDIM)
      __builtin_prefetch(lpb + (size_t)(t0 + 16) * TXDIM + lane * 8, 0, 3);
#pragma unroll
    for (int d = 0; d < 4; ++d) {
      const int t = t0 + d;
      if (t >= TYDIM) break;
      float col[8] = { blo[d].x, blo[d].y, blo[d].z, blo[d].w,
                       bhi[d].x, bhi[d].y, bhi[d].z, bhi[d].w };
      const int tn = t + 4;
      if (tn < TYDIM) {                   // refill this slot 4 iters ahead
        const float4* p = (const float4*)(lpb + (size_t)tn * TXDIM + lane * 8);
        blo[d] = p[0]; bhi[d] = p[1];
      }
      float up = __shfl_up(v[7], 1, 32);  // prev lane's x-1 value
      float sh[8];
      sh[0] = (lane == 0) ? NEGF : up;
#pragma unroll
      for (int k = 1; k < 8; ++k) sh[k] = v[k - 1];
      unsigned int bits = 0;
#pragma unroll
      for (int k = 0; k < 8; ++k) {
        bits |= (sh[k] > v[k] ? 1u : 0u) << k;     // strict >, as reference
        v[k] = col[k] + fmaxf(v[k], sh[k]);
      }
      db[(size_t)t * 32 + lane] = (unsigned char)bits;
    }
  }
}

// ---------------------------------------------------------------------------
// K3: backtrack from (x_len-1, y_len-1).  Stage the 55KB direction plane in
// LDS so the 1728 dependent reads are LDS-latency, then emit xi + durations
// and the per-batch duration-loss partial.
// ---------------------------------------------------------------------------
__global__ void __launch_bounds__(256) gtts_backtrack_kernel(
    const unsigned char* __restrict__ dirs, const float* __restrict__ logw,
    const int* __restrict__ xlen, const int* __restrict__ ylen,
    int* __restrict__ xi, float* __restrict__ acc)
{
  __shared__ unsigned char sdirs[TYDIM * 32];
  __shared__ int sxi[TYDIM];
  __shared__ unsigned int sdur[TXDIM];

  const int b = blockIdx.x, tid = threadIdx.x;
  const unsigned int* src = (const unsigned int*)(dirs + (size_t)b * TYDIM * 32);
  unsigned int* dst = (unsigned int*)sdirs;
  for (int i = tid; i < TYDIM * 32 / 4; i += 256) dst[i] = src[i];
  sdur[tid] = 0u;
  __syncthreads();

  if (tid == 0) {
    const int xl = xlen[b], yl = ylen[b];
    int idx = xl - 1;
    if (idx < 0) idx = 0;
    for (int t = yl - 1; t >= 0; --t) {
      sxi[t] = idx;
      sdur[idx] += 1u;
      if (t > 0) {
        const int byt = sdirs[t * 32 + (idx >> 3)];
        idx -= (byt >> (idx & 7)) & 1;
        if (idx < 0) idx = 0;            // clamp like jnp.take_along_axis
      }
    }
    for (int t = yl; t < TYDIM; ++t) sxi[t] = -1;
  }
  __syncthreads();

  for (int t = tid; t < TYDIM; t += 256) xi[(size_t)b * TYDIM + t] = sxi[t];

  const int xl = xlen[b];
  if (tid < xl) {
    const float lw_ = logf(1e-8f + (float)sdur[tid]);
    const float d = logw[(size_t)b * TXDIM + tid] - lw_;
    atomicAdd(acc + 0, d * d);
  }
}

// ---------------------------------------------------------------------------
// K4a: materialize attn [B][Tx][Ty] (xi == -1 rows are already zero)
// ---------------------------------------------------------------------------
__global__ void __launch_bounds__(256) gtts_attn_kernel(
    const int* __restrict__ xi, float* __restrict__ attn)
{
  const int i = blockIdx.x * 256 + threadIdx.x;        // grid == N/256 exactly
  const int b = i / (TXDIM * TYDIM);
  const int r = i % (TXDIM * TYDIM);
  const int x = r / TYDIM;
  const int t = r % TYDIM;
  attn[i] = (xi[b * TYDIM + t] == x) ? 1.0f : 0.0f;
}

// ---------------------------------------------------------------------------
// K4b: mu_y gather + prior-loss partial (block-reduced, one atomic per block)
// ---------------------------------------------------------------------------
__global__ void __launch_bounds__(256) gtts_muy_prior_kernel(
    const float* __restrict__ mu_x, const float* __restrict__ y,
    const int* __restrict__ xi, float* __restrict__ mu_y,
    float* __restrict__ acc)
{
  const int i = blockIdx.x * 256 + threadIdx.x;        // grid == N/256 exactly
  const int b = i / (TYDIM * NMEL);
  const int r = i % (TYDIM * NMEL);
  const int t = r / NMEL;
  const int m = r % NMEL;
  const int xv = xi[b * TYDIM + t];
  float v = 0.0f, p = 0.0f;
  if (xv >= 0) {
    v = mu_x[((size_t)b * TXDIM + xv) * NMEL + m];
    const float d = y[i] - v;
    p = 0.5f * (d * d + LOG2PI_F);
  }
  mu_y[i] = v;

  __shared__ float sred[256];
  sred[threadIdx.x] = p;
  __syncthreads();
  for (int s = 128; s > 0; s >>= 1) {
    if (threadIdx.x < s) sred[threadIdx.x] += sred[threadIdx.x + s];
    __syncthreads();
  }
  if (threadIdx.x == 0) atomicAdd(acc + 1, sred[0]);
}

// ---------------------------------------------------------------------------
// K5: finalize scalar losses
// ---------------------------------------------------------------------------
__global__ void gtts_finalize_kernel(
    const float* __restrict__ acc, const int* __restrict__ xlen,
    const int* __restrict__ ylen, float* __restrict__ out)
{
  if (threadIdx.x == 0 && blockIdx.x == 0) {
    float sx = 0.0f, sy = 0.0f;
    for (int b = 0; b < BB; ++b) { sx += (float)xlen[b]; sy += (float)ylen[b]; }
    out[0] = acc[0] / sx;                       // dur_loss
    out[1] = acc[1] / (sy * (float)NMEL);       // prior_loss
  }
}

extern "C" void kernel_launch(void* const* d_in, const int* in_sizes, int n_in,
                              void* d_out, int out_size, void* d_ws, size_t ws_size,
                              hipStream_t stream) {
  const float* mu_x = (const float*)d_in[0];
  const float* logw = (const float*)d_in[1];
  const float* y    = (const float*)d_in[2];
  const int*   xlen = (const int*)d_in[3];
  const int*   ylen = (const int*)d_in[4];

  char* ws = (char*)d_ws;
  float*         lp   = (float*)(ws + LP_OFF);
  unsigned char* dirs = (unsigned char*)(ws + DIRS_OFF);
  int*           xi   = (int*)(ws + XI_OFF);
  float*         ysq  = (float*)(ws + YSQ_OFF);
  float*         msq  = (float*)(ws + MSQ_OFF);
  float*         acc  = (float*)(ws + ACC_OFF);

  float* out  = (float*)d_out;
  float* mu_y = out + 2;
  float* attn = out + 2 + (size_t)BB * TYDIM * NMEL;

  gtts_sq_kernel<<<(BB * (TYDIM + TXDIM)) / 256, 256, 0, stream>>>(mu_x, y, ysq, msq, acc);

  dim3 g1(TXDIM / 32, TYDIM / 32, BB);   // 2x2 register-blocked tiles
  gtts_logprior_wmma_kernel<<<g1, 32, 0, stream>>>(mu_x, y, ysq, msq, xlen, ylen, lp);

  gtts_viterbi_fwd_kernel<<<BB, 32, 0, stream>>>(lp, dirs);

  gtts_backtrack_kernel<<<BB, 256, 0, stream>>>(dirs, logw, xlen, ylen, xi, acc);

  gtts_attn_kernel<<<(BB * TXDIM * TYDIM) / 256, 256, 0, stream>>>(xi, attn);

  gtts_muy_prior_kernel<<<(BB * TYDIM * NMEL) / 256, 256, 0, stream>>>(mu_x, y, xi, mu_y, acc);

  gtts_finalize_kernel<<<1, 32, 0, stream>>>(acc, xlen, ylen, out);
}